// HybridTimerGNNBlock_40492951666856
// MI455X (gfx1250) — compile-verified
//
#include <hip/hip_runtime.h>
#include <hip/hip_bf16.h>
#include <math.h>

#define Bc 2
#define Sc 16
#define Nc 5000
#define Hc 64
#define Ec 20000
#define NHEAD 4

typedef __attribute__((ext_vector_type(2))) float v2f;
typedef __attribute__((ext_vector_type(8))) float v8f;

__device__ __forceinline__ float gelu_f(float x) {
  return 0.5f * x * (1.0f + erff(x * 0.70710678118654752f));
}

// ---------------- per-wave 16x16 f32 WMMA tile helpers ----------------
// A fragment: lane L holds row M=L&15, K = k + 2*(L>>4) + {0,1}
// B fragment: lane L holds col N=L&15, K = k + 2*(L>>4) + {0,1}
// C/D fragment: lane L, vgpr i -> row M = i + 8*(L>>4), col N = L&15

__device__ __forceinline__ v8f tile_bias(const float* __restrict__ b, int c0) {
  float bv = b[c0 + (threadIdx.x & 15)];
  v8f c;
#pragma unroll
  for (int i = 0; i < 8; ++i) c[i] = bv;
  return c;
}

// OUT(16x16) += A(16xK in LDS, row-major lda) @ B(KxN in GLOBAL, row-major ldb, col off bcol)
__device__ __forceinline__ v8f tile_mm_lds_glb(const float* As, int lda,
                                               const float* __restrict__ Bg,
                                               int ldb, int bcol, int K, v8f c) {
  const int lane = threadIdx.x & 31;
  const int m = lane & 15;
  const int kh = (lane >> 4) << 1;
#pragma unroll 4
  for (int k = 0; k < K; k += 4) {
    v2f a, b;
    a.x = As[m * lda + k + kh];
    a.y = As[m * lda + k + kh + 1];
    b.x = Bg[(size_t)(k + kh) * ldb + bcol + m];
    b.y = Bg[(size_t)(k + kh + 1) * ldb + bcol + m];
    c = __builtin_amdgcn_wmma_f32_16x16x4_f32(false, a, false, b, (short)0, c,
                                              false, false);
  }
  return c;
}

// same but B in LDS
__device__ __forceinline__ v8f tile_mm_lds_lds(const float* As, int lda,
                                               const float* Bs, int ldb,
                                               int bcol, int K, v8f c) {
  const int lane = threadIdx.x & 31;
  const int m = lane & 15;
  const int kh = (lane >> 4) << 1;
#pragma unroll 4
  for (int k = 0; k < K; k += 4) {
    v2f a, b;
    a.x = As[m * lda + k + kh];
    a.y = As[m * lda + k + kh + 1];
    b.x = Bs[(k + kh) * ldb + bcol + m];
    b.y = Bs[(k + kh + 1) * ldb + bcol + m];
    c = __builtin_amdgcn_wmma_f32_16x16x4_f32(false, a, false, b, (short)0, c,
                                              false, false);
  }
  return c;
}

__device__ __forceinline__ void tile_store(float* Ds, int ldd, int r0, int c0,
                                           v8f c) {
  const int lane = threadIdx.x & 31;
  const int col = lane & 15;
  const int rh = (lane >> 4) << 3;
#pragma unroll
  for (int i = 0; i < 8; ++i) Ds[(r0 + rh + i) * ldd + c0 + col] = c[i];
}

// ---------------- Phase 1: temporal attention per (b,n) ----------------
__global__ __launch_bounds__(128) void k_temporal(
    const float* __restrict__ x, const float* __restrict__ wqkv,
    const float* __restrict__ bqkv, const float* __restrict__ wout,
    const float* __restrict__ bout, const float* __restrict__ tng,
    const float* __restrict__ tnb, float* __restrict__ x1) {
  __shared__ float sX[16 * 64];
  __shared__ float sQKV[16 * 192];
  __shared__ float sP[NHEAD][16 * 16];
  __shared__ float sAO[16 * 64];
  __shared__ float sY[16 * 64];
  __shared__ float sMu[16], sRs[16];

  const int tid = threadIdx.x;
  const int w = tid >> 5;
  const int lane = tid & 31;

  const int p = blockIdx.x;  // 0 .. B*N-1
  const int b = p / Nc, n = p % Nc;
  const size_t rowstride = (size_t)Nc * Hc;
  const float* xb = x + ((size_t)b * Sc * Nc + n) * Hc;

  for (int i = tid; i < 16 * 64; i += 128) {
    int s = i >> 6, cdx = i & 63;
    sX[i] = xb[(size_t)s * rowstride + cdx];
  }
  __syncthreads();

  // QKV = X(16x64) @ Wqkv(64x192) + b : 12 tiles, 3 per wave
  for (int jj = 0; jj < 3; ++jj) {
    int ct = w * 3 + jj;
    v8f c = tile_bias(bqkv, ct * 16);
    c = tile_mm_lds_glb(sX, 64, wqkv, 192, ct * 16, 64, c);
    tile_store(sQKV, 192, 0, ct * 16, c);
  }
  __syncthreads();

  // attention: wave w handles head w (hd = 16)
  {
    const int m = lane & 15;
    const int kh = (lane >> 4) << 1;
    v8f s = {};
#pragma unroll
    for (int k = 0; k < 16; k += 4) {
      v2f a, bb;
      a.x = sQKV[m * 192 + w * 16 + k + kh];        // Q[m][k..]
      a.y = sQKV[m * 192 + w * 16 + k + kh + 1];
      bb.x = sQKV[m * 192 + 64 + w * 16 + k + kh];  // K^T[k..][m] = K[m][k..]
      bb.y = sQKV[m * 192 + 64 + w * 16 + k + kh + 1];
      s = __builtin_amdgcn_wmma_f32_16x16x4_f32(false, a, false, bb, (short)0,
                                                s, false, false);
    }
    // row softmax, scale 1/sqrt(16)=0.25; rows live in 16-lane halves
#pragma unroll
    for (int i = 0; i < 8; ++i) {
      float v = s[i] * 0.25f;
      float mx = v;
      mx = fmaxf(mx, __shfl_xor(mx, 1, 32));
      mx = fmaxf(mx, __shfl_xor(mx, 2, 32));
      mx = fmaxf(mx, __shfl_xor(mx, 4, 32));
      mx = fmaxf(mx, __shfl_xor(mx, 8, 32));
      float e = __expf(v - mx);
      float sm = e;
      sm += __shfl_xor(sm, 1, 32);
      sm += __shfl_xor(sm, 2, 32);
      sm += __shfl_xor(sm, 4, 32);
      sm += __shfl_xor(sm, 8, 32);
      s[i] = e / sm;
    }
    tile_store(sP[w], 16, 0, 0, s);
    // AO_h = P(16x16) @ V(16x16)
    v8f ao = {};
    ao = tile_mm_lds_lds(sP[w], 16, sQKV, 192, 128 + w * 16, 16, ao);
    tile_store(sAO, 64, 0, w * 16, ao);
  }
  __syncthreads();

  // out_proj: AO(16x64) @ Wout(64x64), 4 tiles, one per wave
  {
    v8f c = tile_bias(bout, w * 16);
    c = tile_mm_lds_glb(sAO, 64, wout, 64, w * 16, 64, c);
    tile_store(sY, 64, 0, w * 16, c);
  }
  __syncthreads();

  // LayerNorm(X + Y)
  if (tid < 16) {
    float mu = 0.f;
    for (int c = 0; c < 64; ++c) mu += sX[tid * 64 + c] + sY[tid * 64 + c];
    mu *= (1.f / 64.f);
    float var = 0.f;
    for (int c = 0; c < 64; ++c) {
      float d = sX[tid * 64 + c] + sY[tid * 64 + c] - mu;
      var += d * d;
    }
    sMu[tid] = mu;
    sRs[tid] = rsqrtf(var * (1.f / 64.f) + 1e-5f);
  }
  __syncthreads();
  float* o = x1 + ((size_t)b * Sc * Nc + n) * Hc;
  for (int i = tid; i < 16 * 64; i += 128) {
    int s = i >> 6, cdx = i & 63;
    float v = sX[i] + sY[i];
    o[(size_t)s * rowstride + cdx] = (v - sMu[s]) * sRs[s] * tng[cdx] + tnb[cdx];
  }
}

// ---------------- edge embedding MLP (tiny, VALU) ----------------
__global__ __launch_bounds__(256) void k_edge_emb(
    const float* __restrict__ eattr, const float* __restrict__ ew1,
    const float* __restrict__ eb1, const float* __restrict__ ew2,
    const float* __restrict__ eb2, float* __restrict__ eemb) {
  __shared__ float sH[64 * 64];
  __shared__ float sW2[64 * 64];
  const int tid = threadIdx.x;
  const int e0 = blockIdx.x * 64;
  for (int i = tid; i < 64 * 64; i += 256) sW2[i] = ew2[i];
  for (int i = tid; i < 64 * 64; i += 256) {
    int r = i >> 6, c = i & 63;
    int e = e0 + r;
    float a = (e < Ec) ? eattr[e] : 0.f;
    sH[i] = gelu_f(a * ew1[c] + eb1[c]);
  }
  __syncthreads();
  for (int i = tid; i < 64 * 64; i += 256) {
    int r = i >> 6, c = i & 63;
    int e = e0 + r;
    if (e >= Ec) continue;
    float acc = eb2[c];
    for (int k = 0; k < 64; ++k) acc += sH[r * 64 + k] * sW2[k * 64 + c];
    eemb[(size_t)e * 64 + c] = acc;
  }
}

__global__ void k_zero(float* __restrict__ p, size_t n) {
  size_t i = (size_t)blockIdx.x * blockDim.x + threadIdx.x;
  if (i < n) p[i] = 0.f;
}

// ---------------- message MLP + scatter-add ----------------
__global__ __launch_bounds__(256) void k_message(
    const float* __restrict__ x1, const int* __restrict__ eidx,
    const float* __restrict__ eemb, const float* __restrict__ mw1,
    const float* __restrict__ mb1, const float* __restrict__ mw2,
    const float* __restrict__ mb2, float* __restrict__ aggr) {
  __shared__ float sIn[32 * 192];
  __shared__ float sH[32 * 128];
  __shared__ float sM[32 * 64];
  __shared__ int sSrc[32], sDst[32], sT[32], sE[32];
  const int tid = threadIdx.x;
  const int w = tid >> 5;
  const long g = blockIdx.x;  // rows g*32 .. g*32+31 of (B*S*E)

  if (tid < 32) {
    long rg = g * 32 + tid;
    int t = (int)(rg / Ec);
    int e = (int)(rg % Ec);
    sT[tid] = t;
    sE[tid] = e;
    sSrc[tid] = eidx[e];
    sDst[tid] = eidx[Ec + e];
  }
  __syncthreads();
  for (int i = tid; i < 32 * 192; i += 256) {
    int r = i / 192, c = i % 192;
    size_t base = (size_t)sT[r] * Nc * 64;
    float v;
    if (c < 64) {
      v = x1[base + (size_t)sSrc[r] * 64 + c];
    } else if (c < 128) {
      int h = c - 64;
      v = x1[base + (size_t)sDst[r] * 64 + h] -
          x1[base + (size_t)sSrc[r] * 64 + h];
    } else {
      v = eemb[(size_t)sE[r] * 64 + (c - 128)];
    }
    sIn[i] = v;
  }
  __syncthreads();
  // GEMM1: (32x192)@(192x128)+b -> gelu -> sH ; 16 tiles, 2 per wave
  for (int jj = 0; jj < 2; ++jj) {
    int id = w * 2 + jj;
    int rt = id >> 3, ct = id & 7;
    v8f c = tile_bias(mb1, ct * 16);
    c = tile_mm_lds_glb(sIn + rt * 16 * 192, 192, mw1, 128, ct * 16, 192, c);
#pragma unroll
    for (int i = 0; i < 8; ++i) c[i] = gelu_f(c[i]);
    tile_store(sH, 128, rt * 16, ct * 16, c);
  }
  __syncthreads();
  // GEMM2: (32x128)@(128x64)+b -> sM ; 8 tiles, 1 per wave
  {
    int rt = w >> 2, ct = w & 3;
    v8f c = tile_bias(mb2, ct * 16);
    c = tile_mm_lds_glb(sH + rt * 16 * 128, 128, mw2, 64, ct * 16, 128, c);
    tile_store(sM, 64, rt * 16, ct * 16, c);
  }
  __syncthreads();
  for (int i = tid; i < 32 * 64; i += 256) {
    int r = i >> 6, c = i & 63;
    atomicAdd(&aggr[((size_t)sT[r] * Nc + sDst[r]) * 64 + c], sM[i]);
  }
}

// ---------------- node update MLP + spatial LN (in-place x1) ----------------
__global__ __launch_bounds__(256) void k_update(
    float* __restrict__ x1, const float* __restrict__ aggr,
    const float* __restrict__ uw1, const float* __restrict__ ub1,
    const float* __restrict__ uw2, const float* __restrict__ ub2,
    const float* __restrict__ sng, const float* __restrict__ snb) {
  __shared__ float sIn[32 * 128];
  __shared__ float sG[32 * 64];
  __shared__ float sO[32 * 64];
  __shared__ float sMu[32], sRs[32];
  const int tid = threadIdx.x;
  const int w = tid >> 5;
  const size_t r0 = (size_t)blockIdx.x * 32;  // rows over B*S*N

  for (int i = tid; i < 32 * 128; i += 256) {
    int r = i >> 7, c = i & 127;
    size_t row = r0 + r;
    sIn[i] = (c < 64) ? x1[row * 64 + c] : aggr[row * 64 + (c - 64)];
  }
  __syncthreads();
  {
    int rt = w >> 2, ct = w & 3;
    v8f c = tile_bias(ub1, ct * 16);
    c = tile_mm_lds_glb(sIn + rt * 16 * 128, 128, uw1, 64, ct * 16, 128, c);
#pragma unroll
    for (int i = 0; i < 8; ++i) c[i] = gelu_f(c[i]);
    tile_store(sG, 64, rt * 16, ct * 16, c);
  }
  __syncthreads();
  {
    int rt = w >> 2, ct = w & 3;
    v8f c = tile_bias(ub2, ct * 16);
    c = tile_mm_lds_glb(sG + rt * 16 * 64, 64, uw2, 64, ct * 16, 64, c);
    tile_store(sO, 64, rt * 16, ct * 16, c);
  }
  __syncthreads();
  if (tid < 32) {
    float mu = 0.f;
    for (int c = 0; c < 64; ++c) mu += sIn[tid * 128 + c] + sO[tid * 64 + c];
    mu *= (1.f / 64.f);
    float var = 0.f;
    for (int c = 0; c < 64; ++c) {
      float d = sIn[tid * 128 + c] + sO[tid * 64 + c] - mu;
      var += d * d;
    }
    sMu[tid] = mu;
    sRs[tid] = rsqrtf(var * (1.f / 64.f) + 1e-5f);
  }
  __syncthreads();
  for (int i = tid; i < 32 * 64; i += 256) {
    int r = i >> 6, c = i & 63;
    float v = sIn[r * 128 + c] + sO[i];
    x1[(r0 + r) * 64 + c] = (v - sMu[r]) * sRs[r] * sng[c] + snb[c];
  }
}

// ---------------- FFN + final LN -> d_out ----------------
__global__ __launch_bounds__(256) void k_ffn(
    const float* __restrict__ x1, const float* __restrict__ fw1,
    const float* __restrict__ fb1, const float* __restrict__ fw2,
    const float* __restrict__ fb2, const float* __restrict__ fng,
    const float* __restrict__ fnb, float* __restrict__ out) {
  __shared__ float sX[32 * 64];
  __shared__ float sHh[32 * 256];
  __shared__ float sO[32 * 64];
  __shared__ float sMu[32], sRs[32];
  const int tid = threadIdx.x;
  const int w = tid >> 5;
  const size_t r0 = (size_t)blockIdx.x * 32;

  for (int i = tid; i < 32 * 64; i += 256) sX[i] = x1[r0 * 64 + i];
  __syncthreads();
  // GEMM1: (32x64)@(64x256)+b -> gelu ; 32 tiles, 4 per wave
  for (int jj = 0; jj < 4; ++jj) {
    int id = w * 4 + jj;
    int rt = id >> 4, ct = id & 15;
    v8f c = tile_bias(fb1, ct * 16);
    c = tile_mm_lds_glb(sX + rt * 16 * 64, 64, fw1, 256, ct * 16, 64, c);
#pragma unroll
    for (int i = 0; i < 8; ++i) c[i] = gelu_f(c[i]);
    tile_store(sHh, 256, rt * 16, ct * 16, c);
  }
  __syncthreads();
  // GEMM2: (32x256)@(256x64)+b ; 8 tiles, 1 per wave
  {
    int rt = w >> 2, ct = w & 3;
    v8f c = tile_bias(fb2, ct * 16);
    c = tile_mm_lds_glb(sHh + rt * 16 * 256, 256, fw2, 64, ct * 16, 256, c);
    tile_store(sO, 64, rt * 16, ct * 16, c);
  }
  __syncthreads();
  if (tid < 32) {
    float mu = 0.f;
    for (int c = 0; c < 64; ++c) mu += sX[tid * 64 + c] + sO[tid * 64 + c];
    mu *= (1.f / 64.f);
    float var = 0.f;
    for (int c = 0; c < 64; ++c) {
      float d = sX[tid * 64 + c] + sO[tid * 64 + c] - mu;
      var += d * d;
    }
    sMu[tid] = mu;
    sRs[tid] = rsqrtf(var * (1.f / 64.f) + 1e-5f);
  }
  __syncthreads();
  for (int i = tid; i < 32 * 64; i += 256) {
    int r = i >> 6, c = i & 63;
    float v = sX[i] + sO[i];
    out[r0 * 64 + i] = (v - sMu[r]) * sRs[r] * fng[c] + fnb[c];
  }
}

extern "C" void kernel_launch(void* const* d_in, const int* in_sizes, int n_in,
                              void* d_out, int out_size, void* d_ws,
                              size_t ws_size, hipStream_t stream) {
  (void)in_sizes; (void)n_in; (void)out_size; (void)ws_size;
  const float* x    = (const float*)d_in[0];
  const int*   eidx = (const int*)d_in[1];
  const float* eatt = (const float*)d_in[2];
  const float* wqkv = (const float*)d_in[3];
  const float* bqkv = (const float*)d_in[4];
  const float* wout = (const float*)d_in[5];
  const float* bout = (const float*)d_in[6];
  const float* tng  = (const float*)d_in[7];
  const float* tnb  = (const float*)d_in[8];
  const float* ew1  = (const float*)d_in[9];
  const float* eb1  = (const float*)d_in[10];
  const float* ew2  = (const float*)d_in[11];
  const float* eb2  = (const float*)d_in[12];
  const float* mw1  = (const float*)d_in[13];
  const float* mb1  = (const float*)d_in[14];
  const float* mw2  = (const float*)d_in[15];
  const float* mb2  = (const float*)d_in[16];
  const float* uw1  = (const float*)d_in[17];
  const float* ub1  = (const float*)d_in[18];
  const float* uw2  = (const float*)d_in[19];
  const float* ub2  = (const float*)d_in[20];
  const float* sng  = (const float*)d_in[21];
  const float* snb  = (const float*)d_in[22];
  const float* fw1  = (const float*)d_in[23];
  const float* fb1  = (const float*)d_in[24];
  const float* fw2  = (const float*)d_in[25];
  const float* fb2  = (const float*)d_in[26];
  const float* fng  = (const float*)d_in[27];
  const float* fnb  = (const float*)d_in[28];

  const size_t nelem = (size_t)Bc * Sc * Nc * Hc;  // 10,240,000
  float* x1   = (float*)d_ws;       // phase-1/2 activations, (B,S,N,H)
  float* aggr = x1 + nelem;         // scatter accumulator
  float* eemb = aggr + nelem;       // edge embeddings (E x 64)
  // total workspace: ~87 MB

  k_temporal<<<Bc * Nc, 128, 0, stream>>>(x, wqkv, bqkv, wout, bout, tng, tnb, x1);
  k_edge_emb<<<(Ec + 63) / 64, 256, 0, stream>>>(eatt, ew1, eb1, ew2, eb2, eemb);
  k_zero<<<(unsigned)((nelem + 255) / 256), 256, 0, stream>>>(aggr, nelem);
  k_message<<<(Bc * Sc * Ec) / 32, 256, 0, stream>>>(x1, eidx, eemb, mw1, mb1,
                                                     mw2, mb2, aggr);
  k_update<<<(Bc * Sc * Nc) / 32, 256, 0, stream>>>(x1, aggr, uw1, ub1, uw2,
                                                    ub2, sng, snb);
  k_ffn<<<(Bc * Sc * Nc) / 32, 256, 0, stream>>>(x1, fw1, fb1, fw2, fb2, fng,
                                                 fnb, (float*)d_out);
}